// SigJoinLayer_87333864997629
// MI455X (gfx1250) — compile-verified
//
#include <hip/hip_runtime.h>
#include <hip/hip_bf16.h>

// Sig-join (Chen's identity), M=4, D=8, B=16384.
// Memory-bound streaming kernel (~613 MB total traffic, ~0.3 GFLOP):
// one 256-thread block per row; small per-row tensors in LDS; the rank-1
// x2 (x) p2 (64x64) term computed with V_WMMA_F32_16X16X4_F32 (16 tiles,
// 2 per wave) into LDS so the XDL pipe co-executes with the VALU stream.

typedef __attribute__((ext_vector_type(2))) float v2f;
typedef __attribute__((ext_vector_type(8))) float v8f;

#define SIG   4680   // 8 + 64 + 512 + 4096
#define OFF2  8
#define OFF3  72
#define OFF4  584

__global__ __launch_bounds__(256) void sigjoin_kernel(
    const float* __restrict__ x, const float* __restrict__ y,
    float* __restrict__ out)
{
    __shared__ float sy[8];
    __shared__ float sx1[8];
    __shared__ float sx2[64];
    __shared__ float sp2[64];
    __shared__ float sp3[512];   // p3[jkl] = p2[jk]*y[l]/3
    __shared__ float sq3[512];   // q3[ijk] = x3[ijk] + p3[ijk]/4
    __shared__ float T[64 * 64]; // T[ij][kl] = x2[ij]*p2[kl]  (via WMMA)

    const int row = blockIdx.x;
    const int tid = threadIdx.x;
    const float* xr = x + (long)row * SIG;
    const float* yr = y + (long)row * 8;
    float* orow = out + (long)row * SIG;

    // ---- Level 1: out1 = x1 + y ----
    if (tid < 8) {
        float yv  = yr[tid];
        float x1v = xr[tid];
        sy[tid]  = yv;
        sx1[tid] = x1v;
        orow[tid] = x1v + yv;
    }
    __syncthreads();

    // ---- Level 2: p2 = y(x)y/2 ; out2 = x2 + x1(x)y + p2 ----
    if (tid < 64) {
        int i = tid >> 3, j = tid & 7;
        float x2v = xr[OFF2 + tid];
        float p2v = 0.5f * sy[i] * sy[j];
        sx2[tid] = x2v;
        sp2[tid] = p2v;
        orow[OFF2 + tid] = x2v + sx1[i] * sy[j] + p2v;
    }
    __syncthreads();

    // ---- Level 3: p3 = p2(x)y/3 ; q3 = x3 + p3/4 ;
    //      out3 = x3 + x2(x)y + x1(x)p2 + p3 ----
    #pragma unroll
    for (int e = tid; e < 512; e += 256) {
        float x3v = xr[OFF3 + e];
        float p3v = (1.0f / 3.0f) * sp2[e >> 3] * sy[e & 7];
        sp3[e] = p3v;
        sq3[e] = x3v + 0.25f * p3v;
        orow[OFF3 + e] =
            x3v + sx2[e >> 3] * sy[e & 7] + sx1[e >> 6] * sp2[e & 63] + p3v;
    }

    // ---- WMMA: T = x2 (x) p2 as 64x64, 16 tiles of 16x16, K=4 (col/row 0
    //      carry the vectors, rest zero). 8 waves x 2 tiles. EXEC all-ones.
    {
        int wave = tid >> 5;
        int lane = tid & 31;
        #pragma unroll
        for (int t = wave * 2; t < wave * 2 + 2; ++t) {
            int r0 = (t >> 2) << 4;   // row-block of 64x64 (ij)
            int c0 = (t & 3) << 4;    // col-block of 64x64 (kl)
            float av = sx2[r0 + (lane & 15)];
            float bv = sp2[c0 + (lane & 15)];
            v2f a, b;
            a.x = (lane < 16) ? av : 0.0f;  // A[M][K=0]; lanes>=16 are K=2 -> 0
            a.y = 0.0f;                      // K=1 / K=3 -> 0
            b.x = (lane < 16) ? bv : 0.0f;  // B[K=0][N]; lanes>=16 are K=2 -> 0
            b.y = 0.0f;
            v8f c = {};
            v8f d = __builtin_amdgcn_wmma_f32_16x16x4_f32(
                false, a, false, b, (short)0, c, false, false);
            int mbase = r0 + ((lane >= 16) ? 8 : 0);
            int n = c0 + (lane & 15);
            #pragma unroll
            for (int v = 0; v < 8; ++v)
                T[(mbase + v) * 64 + n] = d[v];
        }
    }
    __syncthreads();

    // ---- Level 4 stream: out4 = x4 + q3(x)y + T + x1(x)p3
    //      float4 loads/stores, fully coalesced (consecutive tid -> consecutive
    //      16B chunks). 4 chunks per thread = 4096 floats per row. ----
    const float4* x4v = reinterpret_cast<const float4*>(xr + OFF4);
    float4* o4v = reinterpret_cast<float4*>(orow + OFF4);
    #pragma unroll
    for (int q = 0; q < 4; ++q) {
        int ch   = q * 256 + tid;   // 0..1023
        int pos  = ch << 2;         // element offset within level 4
        int l0   = pos & 7;         // 0 or 4
        int idx3 = pos >> 3;        // ijk in 0..511
        int r    = idx3 >> 3;       // ij
        int kk   = idx3 & 7;        // k
        float q3v = sq3[idx3];
        float x1v = sx1[idx3 >> 6];
        const float* Trow = &T[r * 64 + kk * 8];
        const float* p3row = &sp3[(idx3 & 63) * 8];
        float4 xin = x4v[ch];
        float4 o;
        float* op = &o.x;
        const float* xp = &xin.x;
        #pragma unroll
        for (int s = 0; s < 4; ++s) {
            int l = l0 + s;
            op[s] = xp[s] + q3v * sy[l] + Trow[l] + x1v * p3row[l];
        }
        o4v[ch] = o;
    }
}

extern "C" void kernel_launch(void* const* d_in, const int* in_sizes, int n_in,
                              void* d_out, int out_size, void* d_ws, size_t ws_size,
                              hipStream_t stream) {
    const float* x = (const float*)d_in[0];
    const float* y = (const float*)d_in[1];
    float* out = (float*)d_out;
    int rows = in_sizes[1] / 8;   // 16384
    sigjoin_kernel<<<rows, 256, 0, stream>>>(x, y, out);
}